// NoisyTopkRouter_44822278701273
// MI455X (gfx1250) — compile-verified
//
#include <hip/hip_runtime.h>

typedef __attribute__((ext_vector_type(2))) float v2f;
typedef __attribute__((ext_vector_type(8))) float v8f;

#define D_DIM 4096
#define E_DIM 64
#define KC 32                 // K rows staged per chunk
#define TPB 128               // tokens per block (8 waves x 16)
#define PSTRIDE 67            // probs LDS stride, gcd(67,64)=1 -> conflict-free
#define WBUF_F 4096           // [2 mats][16 kp][64 n][2 k01] swizzled W chunk
#define NBUF_F 4096           // float offset of noise buffer (128 x 64)
#define PBUF_F 12288          // float offset of probs buffer
#define SMEM_F (PBUF_F + TPB * PSTRIDE)
#define NEG_INF -3.402823466e38f

__global__ __launch_bounds__(256) void router_kernel(
    const float* __restrict__ x, const float* __restrict__ noise,
    const float* __restrict__ Wr, const float* __restrict__ br,
    const float* __restrict__ Wn, const float* __restrict__ bn,
    float* __restrict__ out_w, int* __restrict__ out_idx,
    float* __restrict__ out_p)
{
    __shared__ __align__(16) float smem[SMEM_F];

    const int tid  = threadIdx.x;
    const int wave = tid >> 5;
    const int lane = tid & 31;
    const int col  = lane & 15;
    const int half = lane >> 4;

    const int block_tok0 = blockIdx.x * TPB;
    const int tok0       = block_tok0 + wave * 16;

    // ---- async-stage this block's noise strip (32 KB) into LDS; overlaps GEMM
    {
        const uint32_t lds_nbuf = (uint32_t)(uintptr_t)(void*)(smem + NBUF_F);
        const float4*  gsrc     = (const float4*)(noise + (size_t)block_tok0 * E_DIM);
        #pragma unroll
        for (int j = 0; j < 8; ++j) {
            const int      i   = tid + j * 256;            // 2048 float4 total
            const uint32_t dst = lds_nbuf + (uint32_t)i * 16u;
            const float4*  g   = gsrc + i;
            asm volatile("global_load_async_to_lds_b128 %0, %1, off"
                         :: "v"(dst), "v"(g) : "memory");
        }
    }

    // A-matrix 16x4 fp32: lanes 0-15 -> M, K={0,1}; lanes 16-31 -> M, K={2,3}
    const float* aptr = x + (size_t)(tok0 + col) * D_DIM + half * 2;
    // per-lane float offset into swizzled W: pair-interleaved [kp][n][2]
    const int lb = half * 128 + col * 2;

    v8f acc[2][4];
    #pragma unroll
    for (int m = 0; m < 2; ++m)
        #pragma unroll
        for (int t = 0; t < 4; ++t)
            acc[m][t] = (v8f){0.f, 0.f, 0.f, 0.f, 0.f, 0.f, 0.f, 0.f};

    // cooperative W-fill mapping: thread -> (row-pair kp, col-quad nc4)
    const int kp  = tid >> 4;   // 0..15
    const int nc4 = tid & 15;   // 0..15

    for (int k0 = 0; k0 < D_DIM; k0 += KC) {
        __syncthreads();
        // stage W_route/W_noise rows [k0,k0+32) pair-interleaved into LDS
        const float4 ar0 = ((const float4*)(Wr + (size_t)(k0 + 2 * kp)     * E_DIM))[nc4];
        const float4 ar1 = ((const float4*)(Wr + (size_t)(k0 + 2 * kp + 1) * E_DIM))[nc4];
        const float4 an0 = ((const float4*)(Wn + (size_t)(k0 + 2 * kp)     * E_DIM))[nc4];
        const float4 an1 = ((const float4*)(Wn + (size_t)(k0 + 2 * kp + 1) * E_DIM))[nc4];
        float* wdst = smem + kp * 128 + nc4 * 8;
        #pragma unroll
        for (int e = 0; e < 4; ++e) {
            ((v2f*)wdst)[e]          = (v2f){((const float*)&ar0)[e], ((const float*)&ar1)[e]};
            ((v2f*)(wdst + 2048))[e] = (v2f){((const float*)&an0)[e], ((const float*)&an1)[e]};
        }
        __syncthreads();

        __builtin_prefetch(aptr + k0 + KC, 0, 0);   // next chunk's A strip

        #pragma unroll
        for (int s = 0; s < KC / 4; ++s) {
            v2f a = *(const v2f*)(aptr + k0 + s * 4);
            #pragma unroll
            for (int m = 0; m < 2; ++m) {
                #pragma unroll
                for (int t = 0; t < 4; ++t) {
                    // one ds_load_b64: {W[k][n], W[k+1][n]} contiguous
                    v2f b = *(const v2f*)(smem + m * 2048 + s * 256 + t * 32 + lb);
                    acc[m][t] = __builtin_amdgcn_wmma_f32_16x16x4_f32(
                        false, a, false, b, (short)0, acc[m][t], false, false);
                }
            }
        }
    }

    // noise strip must have landed before the epilogue reads it
    asm volatile("s_wait_asynccnt 0x0" ::: "memory");
    __syncthreads();

    // --- epilogue: bias, softplus noise mix, softmax over E=64 ---
    float brv[4], bnv[4];
    #pragma unroll
    for (int t = 0; t < 4; ++t) {
        brv[t] = br[t * 16 + col];
        bnv[t] = bn[t * 16 + col];
    }

    #pragma unroll
    for (int r = 0; r < 8; ++r) {
        const int m_row = half * 8 + r;            // C/D layout: VGPR r, half -> M
        const int token = tok0 + m_row;
        const int tl    = wave * 16 + m_row;       // block-local token
        float lg[4];
        float mx = NEG_INF;
        #pragma unroll
        for (int t = 0; t < 4; ++t) {
            float lr = acc[0][t][r] + brv[t];
            float ln = acc[1][t][r] + bnv[t];
            float nz = smem[NBUF_F + tl * E_DIM + t * 16 + col];
            float sp = (ln > 0.f) ? (ln + log1pf(__expf(-ln))) : log1pf(__expf(ln));
            lg[t] = lr + nz * sp;
            mx = fmaxf(mx, lg[t]);
        }
        #pragma unroll
        for (int off = 1; off <= 8; off <<= 1)
            mx = fmaxf(mx, __shfl_xor(mx, off, 32));

        float ev[4], sum = 0.f;
        #pragma unroll
        for (int t = 0; t < 4; ++t) { ev[t] = __expf(lg[t] - mx); sum += ev[t]; }
        #pragma unroll
        for (int off = 1; off <= 8; off <<= 1)
            sum += __shfl_xor(sum, off, 32);

        const float inv = 1.0f / sum;
        #pragma unroll
        for (int t = 0; t < 4; ++t) {
            float p = ev[t] * inv;
            out_p[(size_t)token * E_DIM + t * 16 + col] = p;
            smem[PBUF_F + tl * PSTRIDE + t * 16 + col] = p;
        }
    }

    __syncthreads();

    // --- top-8 select + renormalize: one thread per token ---
    if (tid < TPB) {
        const int    token = block_tok0 + tid;
        const float* pr    = smem + PBUF_F + tid * PSTRIDE;
        float tv[8]; int ti[8];
        #pragma unroll
        for (int j = 0; j < 8; ++j) { tv[j] = NEG_INF; ti[j] = 0; }
        for (int e = 0; e < E_DIM; ++e) {
            float p = pr[e]; int ie = e;
            #pragma unroll
            for (int j = 0; j < 8; ++j) {   // stable bubble-through insertion
                if (p > tv[j]) {
                    float fv = tv[j]; int fi = ti[j];
                    tv[j] = p; ti[j] = ie;
                    p = fv; ie = fi;
                }
            }
        }
        float s = 0.f;
        #pragma unroll
        for (int j = 0; j < 8; ++j) s += tv[j];
        const float inv = 1.0f / s;
        #pragma unroll
        for (int j = 0; j < 8; ++j) {
            out_w[(size_t)token * 8 + j]   = tv[j] * inv;
            out_idx[(size_t)token * 8 + j] = ti[j];
        }
    }
}

extern "C" void kernel_launch(void* const* d_in, const int* in_sizes, int n_in,
                              void* d_out, int out_size, void* d_ws, size_t ws_size,
                              hipStream_t stream) {
    const float* x     = (const float*)d_in[0];
    const float* noise = (const float*)d_in[1];
    const float* Wr    = (const float*)d_in[2];
    const float* br    = (const float*)d_in[3];
    const float* Wn    = (const float*)d_in[4];
    const float* bn    = (const float*)d_in[5];

    const int tokens = 4 * 4096;               // B*S = 16384
    float* out     = (float*)d_out;
    float* out_w   = out;                      // [B,S,8]
    int*   out_idx = (int*)(out + tokens * 8); // [B,S,8] int32
    float* out_p   = out + 2 * tokens * 8;     // [B,S,64]

    dim3 grid(tokens / TPB), block(256);
    hipLaunchKernelGGL(router_kernel, grid, block, 0, stream,
                       x, noise, Wr, br, Wn, bn, out_w, out_idx, out_p);
}